// GATv2Layer_46411416600713
// MI455X (gfx1250) — compile-verified
//
#include <hip/hip_runtime.h>

#define N_NODES 20000
#define E_EDGES 320000
#define IN_F    128
#define HEADS   8
#define OUT_F   32
#define PROJ    256    // HEADS*OUT_F
#define PROJ3   768    // 3 projections concatenated
#define NEG_SLOPE 0.2f

typedef __attribute__((ext_vector_type(2))) float v2f;
typedef __attribute__((ext_vector_type(8))) float v8f;

// Monotonic float<->uint encoding for atomicMax-based segment max.
__device__ __forceinline__ unsigned fenc(float f) {
  int i = __float_as_int(f);
  return (unsigned)(i ^ ((i >> 31) | 0x80000000));
}
__device__ __forceinline__ float fdec(unsigned u) {
  int i = (int)u;
  int m = ((~i) >> 31) | 0x80000000;
  return __int_as_float(i ^ m);
}

// ---------------------------------------------------------------------------
// Kernel 0: zero accumulators. out is re-zeroed every call (harness poisons).
__global__ void init_kernel(unsigned* __restrict__ segmax, float* __restrict__ denom,
                            float* __restrict__ out) {
  int stride = gridDim.x * blockDim.x;
  for (int i = blockIdx.x * blockDim.x + threadIdx.x; i < N_NODES * PROJ; i += stride) {
    out[i] = 0.f;
    if (i < N_NODES * HEADS) { segmax[i] = 0u; denom[i] = 0.f; }
  }
}

// ---------------------------------------------------------------------------
// Kernel 1: fused projection GEMM via V_WMMA_F32_16X16X4_F32 (exact fp32).
// hall[n][0:256)=x@W.T (values), [256:512)=x@W1.T (h_src), [512:768)=x@W2.T (h_dst)
// Block: 256 threads = 8 waves; block owns a 16-row M tile; each wave computes
// six 16x16 N tiles, K swept in 32 steps of 4.
__global__ __launch_bounds__(256) void proj_kernel(const float* __restrict__ x,
    const float* __restrict__ W, const float* __restrict__ W1,
    const float* __restrict__ W2, float* __restrict__ hall) {
  __shared__ float xs[16 * IN_F];
  const int rowbase = blockIdx.x * 16;
  const int tid = threadIdx.x;
  for (int i = tid; i < 16 * IN_F; i += 256)
    xs[i] = x[rowbase * IN_F + i];
  __syncthreads();

  const int lane  = tid & 31;
  const int wv    = tid >> 5;
  const int m     = lane & 15;   // row (A) / col (B) within tile
  const int khalf = lane >> 4;   // 0: K=+0,+1   1: K=+2,+3  (ISA 16x4 f32 layout)

  // Preload full A fragment set: 16 rows x 128 K -> 32 x v2f (reused 6x).
  v2f afrag[32];
#pragma unroll
  for (int kc = 0; kc < 32; ++kc) {
    const int k = kc * 4 + khalf * 2;
    v2f a; a.x = xs[m * IN_F + k]; a.y = xs[m * IN_F + k + 1];
    afrag[kc] = a;
  }

  for (int nt = wv; nt < PROJ3 / 16; nt += 8) {
    const int c0 = nt * 16;
    const int c  = c0 + m;
    const float* wrow;
    if (c < PROJ)          wrow = W  + c * IN_F;
    else if (c < 2 * PROJ) wrow = W1 + (c - PROJ) * IN_F;
    else                   wrow = W2 + (c - 2 * PROJ) * IN_F;

    v8f acc = {};
#pragma unroll
    for (int kc = 0; kc < 32; ++kc) {
      const int k = kc * 4 + khalf * 2;
      v2f b; b.x = wrow[k]; b.y = wrow[k + 1];   // B[k][n] = Wrow_n[k]
      acc = __builtin_amdgcn_wmma_f32_16x16x4_f32(
          false, afrag[kc], false, b, (short)0, acc, false, false);
    }
    // C layout: VGPR r -> M = r (lanes 0-15) / r+8 (lanes 16-31), N = lane%16.
#pragma unroll
    for (int r = 0; r < 8; ++r) {
      const int row = rowbase + r + khalf * 8;
      hall[(size_t)row * PROJ3 + c] = acc[r];
    }
  }
}

// ---------------------------------------------------------------------------
// Kernel 2: per-edge GATv2 logits + segment max. One wave per edge; 32 lanes
// sweep the 32 feats of a head (coalesced), wave-reduce, lane 0 commits.
__global__ __launch_bounds__(256) void logits_kernel(const int* __restrict__ ei,
    const float* __restrict__ hall, const float* __restrict__ att,
    float* __restrict__ logits, unsigned* __restrict__ segmax) {
  const int gw   = (blockIdx.x * blockDim.x + threadIdx.x) >> 5;
  const int lane = threadIdx.x & 31;
  const int nw   = (gridDim.x * blockDim.x) >> 5;

  float av[HEADS];
#pragma unroll
  for (int h = 0; h < HEADS; ++h) av[h] = att[h * OUT_F + lane];

  for (int e = gw; e < E_EDGES; e += nw) {
    const int s = ei[e];
    const int d = ei[E_EDGES + e];
    const float* hs = hall + (size_t)s * PROJ3 + PROJ;       // h_src (W1)
    const float* hd = hall + (size_t)d * PROJ3 + 2 * PROJ;   // h_dst (W2)
#pragma unroll
    for (int h = 0; h < HEADS; ++h) {
      float v = hs[h * OUT_F + lane] + hd[h * OUT_F + lane];
      v = v > 0.f ? v : NEG_SLOPE * v;
      v *= av[h];
#pragma unroll
      for (int off = 16; off > 0; off >>= 1) v += __shfl_xor(v, off, 32);
      if (lane == 0) {
        logits[(size_t)e * HEADS + h] = v;
        atomicMax(&segmax[(size_t)d * HEADS + h], fenc(v));
      }
    }
  }
}

// ---------------------------------------------------------------------------
// Kernel 3: stable exp + denominator accumulation (overwrites logits in place).
__global__ void expw_kernel(const int* __restrict__ ei, float* __restrict__ logits,
                            const unsigned* __restrict__ segmax,
                            float* __restrict__ denom) {
  const int i = blockIdx.x * blockDim.x + threadIdx.x;
  if (i >= E_EDGES * HEADS) return;
  const int e = i >> 3, h = i & 7;
  const int d = ei[E_EDGES + e];
  const float mx = fdec(segmax[(size_t)d * HEADS + h]);
  const float w = __expf(logits[i] - mx);
  logits[i] = w;
  atomicAdd(&denom[(size_t)d * HEADS + h], w);
}

// ---------------------------------------------------------------------------
// Kernel 4: weighted scatter of source values into dst rows (f32 atomics,
// L2-resident). Division by denom is hoisted to the finalize kernel.
__global__ __launch_bounds__(256) void agg_kernel(const int* __restrict__ ei,
    const float* __restrict__ hall, const float* __restrict__ wts,
    float* __restrict__ out) {
  const int gw   = (blockIdx.x * blockDim.x + threadIdx.x) >> 5;
  const int lane = threadIdx.x & 31;
  const int nw   = (gridDim.x * blockDim.x) >> 5;
  for (int e = gw; e < E_EDGES; e += nw) {
    const int s = ei[e];
    const int d = ei[E_EDGES + e];
    const float* vals = hall + (size_t)s * PROJ3;   // values (W)
    float* orow = out + (size_t)d * PROJ;
#pragma unroll
    for (int h = 0; h < HEADS; ++h) {
      const float w = wts[(size_t)e * HEADS + h];
      atomicAdd(&orow[h * OUT_F + lane], w * vals[h * OUT_F + lane]);
    }
  }
}

// ---------------------------------------------------------------------------
// Kernel 5: out = accum / denom + bias (guard isolated nodes: denom == 0).
__global__ void final_kernel(float* __restrict__ out, const float* __restrict__ denom,
                             const float* __restrict__ bias) {
  const int i = blockIdx.x * blockDim.x + threadIdx.x;
  if (i >= N_NODES * PROJ) return;
  const int n = i / PROJ, c = i % PROJ;
  const float dn = denom[(size_t)n * HEADS + c / OUT_F];
  const float r = dn > 0.f ? out[i] / dn : 0.f;
  out[i] = r + bias[c];
}

// ---------------------------------------------------------------------------
extern "C" void kernel_launch(void* const* d_in, const int* in_sizes, int n_in,
                              void* d_out, int out_size, void* d_ws, size_t ws_size,
                              hipStream_t stream) {
  const float* x    = (const float*)d_in[0];
  const int*   ei   = (const int*)d_in[1];
  const float* W    = (const float*)d_in[2];
  const float* W1   = (const float*)d_in[3];
  const float* W2   = (const float*)d_in[4];
  const float* att  = (const float*)d_in[5];
  const float* bias = (const float*)d_in[6];
  float* out = (float*)d_out;

  float*    hall   = (float*)d_ws;                                   // N*768
  float*    logits = hall + (size_t)N_NODES * PROJ3;                 // E*8
  unsigned* segmax = (unsigned*)(logits + (size_t)E_EDGES * HEADS);  // N*8
  float*    denom  = (float*)(segmax + (size_t)N_NODES * HEADS);     // N*8

  init_kernel<<<(N_NODES * PROJ + 255) / 256, 256, 0, stream>>>(segmax, denom, out);
  proj_kernel<<<N_NODES / 16, 256, 0, stream>>>(x, W, W1, W2, hall);
  logits_kernel<<<2048, 256, 0, stream>>>(ei, hall, att, logits, segmax);
  expw_kernel<<<(E_EDGES * HEADS + 255) / 256, 256, 0, stream>>>(ei, logits, segmax, denom);
  agg_kernel<<<2048, 256, 0, stream>>>(ei, hall, logits, out);
  final_kernel<<<(N_NODES * PROJ + 255) / 256, 256, 0, stream>>>(out, denom, bias);
}